// Harmonic_81346680586294
// MI455X (gfx1250) — compile-verified
//
#include <hip/hip_runtime.h>
#include <math.h>

typedef __attribute__((ext_vector_type(2))) float v2f;
typedef __attribute__((ext_vector_type(8))) float v8f;

#define CH 64
#define NFRAMES 128
#define NHARM 32
#define NSAMP 32768
#define STRA 130   // bufA row stride (1 pad + 128 + 1 pad)
#define STRB 66    // bufB row stride (1 pad + 64 + 1 pad)

struct NetP {
  const float* iw; const float* ib;        // initial (256,128),(256)
  const float* fw; const float* fb;        // final (1,64,3),(1)
  const float* w1[5]; const float* b1[5];  // per block
  const float* w2[5]; const float* b2[5];
  int layers;
};
struct AllP {
  NetP n[4];
  const float* x;
  float* env; float* f0; float* hm; float* am; double* q;
};

// Conv1d(64->64, k=3, pad=1) as GEMM via V_WMMA_F32_16X16X4_F32.
// A: rows = output positions (16-tile), cols k = t*64+ci (K=192, 48 steps of 4)
// B: W[co][ci][t] for 16 co per wave. UP fuses nearest-x2 upsample into A reads.
template<bool UP>
__device__ __forceinline__ void conv3x(const float* __restrict__ w,
                                       const float* __restrict__ bia,
                                       const float* __restrict__ in, int instr,
                                       float* __restrict__ out, int outstr,
                                       int Lout, int lane, int wave)
{
  const int cobase = wave * 16;
  const int n = lane & 15;
  const int khalf = (lane >> 4) << 1;  // lanes 0-15: K{0,1}, lanes 16-31: K{2,3}
  const float* wrow = w + (cobase + n) * 192;

  v2f bf[48];
#pragma unroll
  for (int s = 0; s < 48; ++s) {
    int k0 = 4 * s + khalf, k1 = k0 + 1;
    int ci0 = k0 & 63, t0 = k0 >> 6;
    int ci1 = k1 & 63, t1 = k1 >> 6;
    bf[s].x = wrow[ci0 * 3 + t0];
    bf[s].y = wrow[ci1 * 3 + t1];
  }
  const float bias = bia[cobase + n];
  const int nMt = (Lout + 15) >> 4;
  for (int mt = 0; mt < nMt; ++mt) {
    const int p = mt * 16 + n;  // A-row (position) owned by this lane
    v8f acc = { bias, bias, bias, bias, bias, bias, bias, bias };
#pragma unroll
    for (int s = 0; s < 48; ++s) {
      int k0 = 4 * s + khalf, k1 = k0 + 1;
      int ci0 = k0 & 63, t0 = k0 >> 6;
      int ci1 = k1 & 63, t1 = k1 >> 6;
      int q0 = p + t0 - 1;
      int q1 = p + t1 - 1;
      if (UP) { q0 >>= 1; q1 >>= 1; }   // -1>>1 == -1 -> hits zero pad at row offset 0
      v2f a;
      a.x = in[ci0 * instr + 1 + q0];
      a.y = in[ci1 * instr + 1 + q1];
      acc = __builtin_amdgcn_wmma_f32_16x16x4_f32(false, a, false, bf[s],
                                                  (short)0, acc, false, false);
    }
    const int rb = (lane < 16) ? 0 : 8;
#pragma unroll
    for (int j = 0; j < 8; ++j) {
      int pp = mt * 16 + rb + j;
      if (pp < Lout) out[(cobase + n) * outstr + 1 + pp] = acc[j];
    }
  }
}

__global__ __launch_bounds__(128) void net_kernel(AllP P) {
  const int a   = blockIdx.x;   // atom 0..31
  const int net = blockIdx.y;   // 0=env 1=f0 2=harm 3=amp
  const NetP& np = P.n[net];
  const int layers = np.layers;
  const int tid  = threadIdx.x;
  const int lane = tid & 31;
  const int wave = tid >> 5;

  __shared__ float bufA[CH * STRA];  // conv2 outputs / conv1 inputs (L<=128)
  __shared__ float bufB[CH * STRB];  // conv1 outputs (L<=64)
  __shared__ float fr[128];          // x staging, later f0 frames

  for (int i = tid; i < CH * STRA; i += 128) bufA[i] = 0.f;
  for (int i = tid; i < CH * STRB; i += 128) bufB[i] = 0.f;
  fr[tid] = P.x[a * 128 + tid];
  __syncthreads();

  // initial 1x1 conv: 128 -> 256, reshaped to (64 ch, 4 pos)
#pragma unroll
  for (int r = 0; r < 2; ++r) {
    const int c = tid + r * 128;
    const float* wr = np.iw + c * 128;
    float acc = np.ib[c];
    for (int ci = 0; ci < 128; ++ci) acc = fmaf(fr[ci], wr[ci], acc);
    bufA[(c >> 2) * STRA + 1 + (c & 3)] = acc;
  }
  __syncthreads();

  int L = 4;
  for (int blk = 0; blk < layers; ++blk) {
    conv3x<false>(np.w1[blk], np.b1[blk], bufA, STRA, bufB, STRB, L, lane, wave);
    __syncthreads();
    conv3x<true >(np.w2[blk], np.b2[blk], bufB, STRB, bufA, STRA, 2 * L, lane, wave);
    __syncthreads();
    L *= 2;
    for (int i = tid; i < CH * L; i += 128) {       // LeakyReLU(0.2) in place
      int ci = i / L, p = i - ci * L;
      float v = bufA[ci * STRA + 1 + p];
      bufA[ci * STRA + 1 + p] = (v >= 0.f) ? v : 0.2f * v;
    }
    __syncthreads();
  }

  // final conv 64 -> 1 (k=3, pad=1) + clip + per-net affine
  if (tid < L) {
    float acc = np.fb[0];
    for (int ci = 0; ci < CH; ++ci) {
      const float* wr = np.fw + ci * 3;
      const float* xr = bufA + ci * STRA + tid;     // positions tid-1..tid+1
      acc = fmaf(xr[0], wr[0], acc);
      acc = fmaf(xr[1], wr[1], acc);
      acc = fmaf(xr[2], wr[2], acc);
    }
    float c01 = fminf(fmaxf(acc, 0.f), 1.f);
    if (net == 0) {
      P.env[a * NFRAMES + tid] = c01;
    } else if (net == 1) {
      float fv = 0.0018140589569161f + c01 * 0.7238095238095238f; // MIN_F0 + c*F0_DIFF
      P.f0[a * NFRAMES + tid] = fv;
      fr[tid] = fv;
    } else if (net == 2) {
      P.hm[a * NHARM + tid] = 1.0f + c01 * 10.0f;
    } else {
      P.am[a * NHARM + tid] = c01;
    }
  }
  __syncthreads();

  // f64 frame-level phase prefix: Q[k] = sum of f0_u over samples [0, 128+256k)
  if (net == 1 && tid == 0) {
    double* Q = P.q + a * NFRAMES;
    double q = 128.0 * (double)fr[0];
    Q[0] = q;
    for (int k = 1; k < 128; ++k) {
      q += 128.0 * ((double)fr[k - 1] + (double)fr[k]);
      Q[k] = q;
    }
  }
}

__device__ __forceinline__ float sinred(double ph) {
  double r = ph - 6.283185307179586476925286766559 *
                  rint(ph * 0.159154943091895335768883763373);
  return __sinf((float)r);
}

__global__ __launch_bounds__(256) void synth_kernel(const float* __restrict__ env,
                                                    const float* __restrict__ f0,
                                                    const float* __restrict__ hm,
                                                    const float* __restrict__ am,
                                                    const double* __restrict__ q,
                                                    float* __restrict__ out) {
  const int a = blockIdx.y;
  const int t = blockIdx.x * 256 + threadIdx.x;
  __shared__ float sh[NHARM], sa[NHARM];
  if (threadIdx.x < NHARM) {
    sh[threadIdx.x] = hm[a * NHARM + threadIdx.x];
    sa[threadIdx.x] = am[a * NHARM + threadIdx.x];
  }
  __syncthreads();
  const float*  E  = env + a * NFRAMES;
  const float*  F  = f0  + a * NFRAMES;
  const double* Qa = q   + a * NFRAMES;

  // linear upsample of env (half-pixel centers, edge clamp)
  float pos = (t + 0.5f) * (1.0f / 256.0f) - 0.5f;
  float env_u;
  if (pos <= 0.f)        env_u = E[0];
  else if (pos >= 127.f) env_u = E[127];
  else {
    float fj = floorf(pos);
    int j = (int)fj;
    float w = pos - fj;
    env_u = fmaf(w, E[j + 1] - E[j], E[j]);
  }

  // analytic cumsum of piecewise-linear f0_u (pre-pi), in f64
  double s;
  if (t < 128) {
    s = (double)(t + 1) * (double)F[0];
  } else if (t >= NSAMP - 128) {
    s = Qa[127] + (double)(t - 32639) * (double)F[127];
  } else {
    int k = (t - 128) >> 8;
    int m = (t - 128) & 255;
    double f0k = (double)F[k];
    double d   = (double)F[k + 1] - f0k;
    double mp1 = (double)(m + 1);
    s = Qa[k] + mp1 * f0k + d * mp1 * mp1 * (1.0 / 512.0);
  }
  double phase = s * 3.14159265358979323846;

  float acc = sinred(phase);
#pragma unroll
  for (int h = 0; h < NHARM; ++h)
    acc = fmaf(sa[h], sinred(phase * (double)sh[h]), acc);

  out[a * NSAMP + t] = env_u * acc;
}

extern "C" void kernel_launch(void* const* d_in, const int* in_sizes, int n_in,
                              void* d_out, int out_size, void* d_ws, size_t ws_size,
                              hipStream_t stream) {
  static const int net_layers[4] = {5, 5, 3, 3};
  AllP P{};
  P.x = (const float*)d_in[0];

  if (n_in > 5) {
    // Flattened leaves: classify by unique element counts; robust to flatten order.
    int li = 1;
    for (int nn = 0; nn < 4 && li < n_in; ++nn) {
      NetP& np = P.n[nn];
      np.layers = net_layers[nn];
      const int needw = 2 * np.layers;
      int nw = 0, nb = 0;
      bool iw = false, ib = false, fw = false, fb = false;
      while (li < n_in && !(nw == needw && nb == needw && iw && ib && fw && fb)) {
        const float* p = (const float*)d_in[li];
        switch (in_sizes[li]) {
          case 32768: np.iw = p; iw = true; break;                 // (256,128,1)
          case 256:   np.ib = p; ib = true; break;
          case 192:   np.fw = p; fw = true; break;                 // (1,64,3)
          case 1:     np.fb = p; fb = true; break;
          case 12288:                                              // (64,64,3)
            if ((nw & 1) == 0) np.w1[nw >> 1] = p; else np.w2[nw >> 1] = p;
            ++nw; break;
          case 64:
            if ((nb & 1) == 0) np.b1[nb >> 1] = p; else np.b2[nb >> 1] = p;
            ++nb; break;
          default: break;
        }
        ++li;
      }
    }
  } else {
    // One concatenated buffer per dict entry, jax pytree (alphabetical) leaf order:
    // blocks[(w1,b1,w2,b2)...], final_b, final_w, initial_b, initial_w
    for (int nn = 0; nn < 4; ++nn) {
      const float* base = (const float*)d_in[1 + nn];
      NetP& np = P.n[nn];
      np.layers = net_layers[nn];
      size_t off = 0;
      for (int b = 0; b < np.layers; ++b) {
        np.w1[b] = base + off; off += 12288;
        np.b1[b] = base + off; off += 64;
        np.w2[b] = base + off; off += 12288;
        np.b2[b] = base + off; off += 64;
      }
      np.fb = base + off; off += 1;
      np.fw = base + off; off += 192;
      np.ib = base + off; off += 256;
      np.iw = base + off; off += 32768;
    }
  }

  char* ws = (char*)d_ws;
  P.env = (float*) (ws + 0);        // 32*128 f32
  P.f0  = (float*) (ws + 16384);    // 32*128 f32
  P.hm  = (float*) (ws + 32768);    // 32*32  f32
  P.am  = (float*) (ws + 36864);    // 32*32  f32
  P.q   = (double*)(ws + 40960);    // 32*128 f64 (8B aligned)

  net_kernel<<<dim3(32, 4), 128, 0, stream>>>(P);
  synth_kernel<<<dim3(NSAMP / 256, 32), 256, 0, stream>>>(P.env, P.f0, P.hm, P.am,
                                                          P.q, (float*)d_out);
}